// S4Layer_37890201485550
// MI455X (gfx1250) — compile-verified
//
#include <hip/hip_runtime.h>
#include <math.h>

// S4 (DPLR SSM) fused frequency-domain convolution for MI455X / gfx1250.
//   kernel 1: build K[4096] (Cauchy kernel) -> d_ws, stored digit-reversed.
//   kernel 2: per row (4096 rows): async-copy 16KB row -> LDS (ASYNCcnt path),
//             radix-16 FFT (3 stages of 16x16 DFT matmuls on
//             V_WMMA_F32_16X16X4_F32), multiply by K, conjugate, 3 more WMMA
//             stages (inverse via forward-FFT-of-conj), async-store Re/L from
//             LDS. One pass over HBM: memory-bound at ~128 MiB total.

#define LFFT   4096
#define NSTATE 64
#define PI2F   6.28318530717958647692f

typedef __attribute__((ext_vector_type(2))) float v2f;
typedef __attribute__((ext_vector_type(8))) float v8f;
typedef int v4i __attribute__((vector_size(16)));   // 128-bit async payload type

#if __has_builtin(__builtin_amdgcn_global_load_async_to_lds_b128) && \
    __has_builtin(__builtin_amdgcn_global_store_async_from_lds_b32)
#define USE_ASYNC 1
#else
#define USE_ASYNC 0
#endif

#define AS_LDS_V4I(p) ((__attribute__((address_space(3))) v4i*)(p))
#define AS_LDS_I32(p) ((__attribute__((address_space(3))) int*)(p))

__device__ __forceinline__ void wait_async0() {
#if __has_builtin(__builtin_amdgcn_s_wait_asynccnt)
  __builtin_amdgcn_s_wait_asynccnt(0);
#else
  asm volatile("s_wait_asynccnt 0" ::: "memory");
#endif
}

struct cf { float x, y; };
__device__ __forceinline__ cf cmul(cf a, cf b){ return {a.x*b.x - a.y*b.y, a.x*b.y + a.y*b.x}; }
__device__ __forceinline__ cf cadd(cf a, cf b){ return {a.x+b.x, a.y+b.y}; }
__device__ __forceinline__ cf csub(cf a, cf b){ return {a.x-b.x, a.y-b.y}; }
__device__ __forceinline__ cf crcp(cf a){ float d = 1.0f/(a.x*a.x + a.y*a.y); return {a.x*d, -a.y*d}; }

// ---------------------------------------------------------------------------
// Kernel 1: frequency-domain DPLR kernel K (4096 complex), digit-reversed.
// ---------------------------------------------------------------------------
__global__ __launch_bounds__(256)
void s4_cauchy_kernel(const float* __restrict__ Lre, const float* __restrict__ Lim,
                      const float* __restrict__ Pre, const float* __restrict__ Pim,
                      const float* __restrict__ Bre, const float* __restrict__ Bim,
                      const float* __restrict__ Cre, const float* __restrict__ Cim,
                      float2* __restrict__ Kout) {
  int k = blockIdx.x * blockDim.x + threadIdx.x;
  if (k >= LFFT) return;
  float ang = -PI2F * (float)k / (float)LFFT;
  float s, c; __sincosf(ang, &s, &c);
  cf onem = {1.0f - c, -s};
  cf onep = {1.0f + c,  s};
  cf ponep = crcp(onep);
  cf g   = cmul({2000.0f * onem.x, 2000.0f * onem.y}, ponep);  // (2/step)*(1-w)/(1+w)
  cf cf2 = {2.0f * ponep.x, 2.0f * ponep.y};                   // 2/(1+w)
  cf k00 = {0,0}, k01 = {0,0}, k10 = {0,0}, k11 = {0,0};
  for (int n = 0; n < NSTATE; ++n) {
    cf lam = {Lre[n], Lim[n]};
    cf d   = crcp({g.x - lam.x, g.y - lam.y});
    cf cc  = {Cre[n], -Cim[n]};          // conj(C)
    cf Bn  = {Bre[n],  Bim[n]};
    cf Pn  = {Pre[n],  Pim[n]};
    cf Pc  = {Pn.x,   -Pn.y};            // conj(P)
    k00 = cadd(k00, cmul(d, cmul(cc, Bn)));
    k01 = cadd(k01, cmul(d, cmul(cc, Pn)));
    k10 = cadd(k10, cmul(d, cmul(Pc, Bn)));
    k11 = cadd(k11, cmul(d, cmul(Pc, Pn)));
  }
  cf K = cmul(cf2, csub(k00, cmul(cmul(k01, crcp({1.0f + k11.x, k11.y})), k10)));
  int kr = ((k & 15) << 8) | (k & 0xF0) | (k >> 8);   // base-16 digit reversal
  Kout[kr] = make_float2(K.x, K.y);
}

// ---------------------------------------------------------------------------
// Kernel 2: fused FFT * K * IFFT, radix-16 stages on f32 WMMA.
// ---------------------------------------------------------------------------
__device__ __forceinline__ v8f wmma4(v2f a, v2f b, v8f c) {
  // V_WMMA_F32_16X16X4_F32 : D(16x16) += A(16x4) * B(4x16)
  return __builtin_amdgcn_wmma_f32_16x16x4_f32(false, a, false, b, (short)0, c, false, false);
}

// element address within a 4096-point row for stage's 16-point DFT groups:
//   stage 0: x[j + 256*t]                    (t = DFT index, j in [0,256))
//   stage 1: x[(j>>4)*256 + (j&15) + 16*t]
//   stage 2: x[j*16 + t]
template <int STAGE>
__device__ __forceinline__ int st_addr(int t, int j) {
  if (STAGE == 0) return j + (t << 8);
  if (STAGE == 1) return ((j >> 4) << 8) | (t << 4) | (j & 15);
  return (j << 4) | t;
}

template <int STAGE, bool FIN = false>
__device__ __forceinline__ void run_stage(const float* __restrict__ iR, const float* __restrict__ iI,
                                          float* __restrict__ oR, float* __restrict__ oI,
                                          const v2f* Ar, const v2f* Ai, const v2f* An,
                                          int lane, int wave) {
  const int hs = lane >> 4;      // half-wave select (K/M high half)
  const int lr = lane & 15;      // A row (M) / B,C column (N)
  for (int tile = wave; tile < 16; tile += 8) {   // 8 waves cover 16 tiles
    const int col = (tile << 4) + lr;
    v8f cre = {}, cim = {};
#pragma unroll
    for (int kc = 0; kc < 4; ++kc) {
      const int k0 = 4 * kc + 2 * hs;
      const int i0 = st_addr<STAGE>(k0,     col);
      const int i1 = st_addr<STAGE>(k0 + 1, col);
      v2f br, bi;
      br[0] = iR[i0]; br[1] = iR[i1];
      bi[0] = iI[i0]; bi[1] = iI[i1];
      // complex: Cre += Ar*Br - Ai*Bi ; Cim += Ai*Br + Ar*Bi
      cre = wmma4(Ar[kc], br, cre);
      cre = wmma4(An[kc], bi, cre);   // An = -Ai (f32 WMMA has no A/B negate)
      cim = wmma4(Ai[kc], br, cim);
      cim = wmma4(Ar[kc], bi, cim);
    }
#pragma unroll
    for (int v = 0; v < 8; ++v) {
      const int M = v + 8 * hs;       // C/D layout: row = vgpr + 8*(lane>=16)
      float re = cre[v], im = cim[v];
      if (STAGE < 2) {                // inter-stage twiddles
        const float a = (STAGE == 0)
            ? (-(PI2F / 4096.0f) * (float)(col * M))
            : (-(PI2F /  256.0f) * (float)((col & 15) * M));
        float s, c; __sincosf(a, &s, &c);
        const float r2 = re * c - im * s;
        const float i2 = re * s + im * c;
        re = r2; im = i2;
      }
      if (FIN) {                      // fold 1/L so LDS holds finished values
        re *= (1.0f / (float)LFFT);
        im *= (1.0f / (float)LFFT);
      }
      const int o = st_addr<STAGE>(M, col);
      oR[o] = re; oI[o] = im;
    }
  }
}

__global__ __launch_bounds__(256)
void s4_fft_kernel(const float* __restrict__ x, const float2* __restrict__ Kw,
                   float* __restrict__ y) {
  __shared__ float sReA[LFFT], sImA[LFFT], sReB[LFFT], sImB[LFFT];

  const int tid  = threadIdx.x;
  const int lane = tid & 31;
  const int wave = tid >> 5;
  const int hs   = lane >> 4;
  const int lr   = lane & 15;
  const size_t base = (size_t)blockIdx.x * LFFT;

  // Kick off the row staging first: 256 lanes x 4 x 16B = 16 KB, ASYNCcnt path
  // (no VGPR landing zone, overlaps with DFT-matrix setup below).
#if USE_ASYNC
  v4i* xv = (v4i*)(x + base);
#pragma unroll
  for (int i = 0; i < 4; ++i) {
    const int e = tid + (i << 8);
    __builtin_amdgcn_global_load_async_to_lds_b128(xv + e,
                                                   AS_LDS_V4I(&sReA[4 * e]),
                                                   0, 0);
  }
#pragma unroll
  for (int i = 0; i < 4; ++i) {
    const int e = tid + (i << 8);
    sImA[4 * e + 0] = 0.0f; sImA[4 * e + 1] = 0.0f;
    sImA[4 * e + 2] = 0.0f; sImA[4 * e + 3] = 0.0f;
  }
#else
  const float4* xv = (const float4*)(x + base);
#pragma unroll
  for (int i = 0; i < 4; ++i) {
    const int e = tid + (i << 8);
    const float4 v = xv[e];
    sReA[4 * e + 0] = v.x; sReA[4 * e + 1] = v.y;
    sReA[4 * e + 2] = v.z; sReA[4 * e + 3] = v.w;
    sImA[4 * e + 0] = 0.0f; sImA[4 * e + 1] = 0.0f;
    sImA[4 * e + 2] = 0.0f; sImA[4 * e + 3] = 0.0f;
  }
#endif

  // Per-lane 16x16 DFT matrix in WMMA-A layout: D[m][k] = exp(-2*pi*i*m*k/16).
  // A(16x4) chunk kc: vgpr u holds K = 4*kc + 2*hs + u for row M = lr.
  v2f Ar[4], Ai[4], An[4];
#pragma unroll
  for (int kc = 0; kc < 4; ++kc) {
#pragma unroll
    for (int u = 0; u < 2; ++u) {
      const int kk = 4 * kc + 2 * hs + u;
      const float th = (PI2F / 16.0f) * (float)(lr * kk);
      float s, c; __sincosf(th, &s, &c);
      Ar[kc][u] = c;  Ai[kc][u] = -s;  An[kc][u] = s;
    }
  }

#if USE_ASYNC
  wait_async0();
#endif
  __syncthreads();

  // Forward FFT: three radix-16 WMMA stages (DIF, digit-reversed output).
  run_stage<0>(sReA, sImA, sReB, sImB, Ar, Ai, An, lane, wave); __syncthreads();
  run_stage<1>(sReB, sImB, sReA, sImA, Ar, Ai, An, lane, wave); __syncthreads();
  run_stage<2>(sReA, sImA, sReB, sImB, Ar, Ai, An, lane, wave); __syncthreads();

  // Pointwise K (stored digit-reversed to match), then conjugate so the
  // inverse transform reuses the forward pipeline: ifft(z) = conj(fft(conj z))/L.
  for (int p = tid; p < LFFT; p += 256) {
    const float2 kw = Kw[p];
    const float xr = sReB[p], xi = sImB[p];
    sReA[p] =  xr * kw.x - xi * kw.y;
    sImA[p] = -(xr * kw.y + xi * kw.x);
  }
  __syncthreads();

  run_stage<0>(sReA, sImA, sReB, sImB, Ar, Ai, An, lane, wave); __syncthreads();
  run_stage<1>(sReB, sImB, sReA, sImA, Ar, Ai, An, lane, wave); __syncthreads();
  run_stage<2, true>(sReA, sImA, sReB, sImB, Ar, Ai, An, lane, wave); __syncthreads();

  // Re(.) (already scaled by 1/L in the last stage), un-digit-reversed on the
  // LDS side so the global side stays coalesced.
#if USE_ASYNC
  for (int q = tid; q < LFFT; q += 256) {
    const int p = ((q & 15) << 8) | (q & 0xF0) | (q >> 8);
    __builtin_amdgcn_global_store_async_from_lds_b32((int*)(y + base + q),
                                                     AS_LDS_I32(&sReB[p]),
                                                     0, 0);
  }
  wait_async0();
#else
  for (int q = tid; q < LFFT; q += 256) {
    const int p = ((q & 15) << 8) | (q & 0xF0) | (q >> 8);
    y[base + q] = sReB[p];
  }
#endif
}

// ---------------------------------------------------------------------------
extern "C" void kernel_launch(void* const* d_in, const int* in_sizes, int n_in,
                              void* d_out, int out_size, void* d_ws, size_t ws_size,
                              hipStream_t stream) {
  (void)in_sizes; (void)n_in; (void)out_size; (void)ws_size;
  const float* x   = (const float*)d_in[0];
  const float* Lre = (const float*)d_in[1];
  const float* Lim = (const float*)d_in[2];
  const float* Pre = (const float*)d_in[3];
  const float* Pim = (const float*)d_in[4];
  const float* Bre = (const float*)d_in[5];
  const float* Bim = (const float*)d_in[6];
  const float* Cre = (const float*)d_in[7];
  const float* Cim = (const float*)d_in[8];
  float2* Kw = (float2*)d_ws;                       // 4096 * 8 B = 32 KB scratch
  float*  y  = (float*)d_out;

  s4_cauchy_kernel<<<LFFT / 256, 256, 0, stream>>>(Lre, Lim, Pre, Pim, Bre, Bim,
                                                   Cre, Cim, Kw);
  s4_fft_kernel<<<16 * 256, 256, 0, stream>>>(x, Kw, y);
}